// PolymerGNN_IV_EXPLAIN_39831526703690
// MI455X (gfx1250) — compile-verified
//
#include <hip/hip_runtime.h>
#include <math.h>

// PolymerGNN forward for MI455X (gfx1250, wave32).
// - GEMMs: V_WMMA_F32_16X16X4_F32 (fp32-exact vs reference; the workload is
//   gather/latency bound at ~12.9 GFLOP, so low-precision WMMA buys nothing).
// - A-operand staged global->LDS with GLOBAL_LOAD_ASYNC_TO_LDS_B128
//   (ASYNCcnt) to kill the 8x redundant per-wave A loads; WMMA A-fragments
//   then come from LDS (ds_load_b64, bank-conflict-free via 132-float row
//   stride).
// - Segment ops exploit dst = repeat(arange(N), DEG): node n owns edges
//   [8n,8n+8) + self loop -> no atomics for GAT/SAGE/score aggregation.

typedef float v2f __attribute__((ext_vector_type(2)));
typedef float v8f __attribute__((ext_vector_type(8)));

namespace {
constexpr int kB     = 64;
constexpr int kNPer  = 1024;
constexpr int kN     = kB * kNPer;   // 65536
constexpr int kDeg   = 8;
constexpr int kF     = 128;
constexpr int kH     = 128;
constexpr int kKPool = kNPer / 2;    // 512
constexpr int kLdsStride = 132;      // padded row stride (floats): conflict-free
constexpr float kEps = 1e-5f;
constexpr unsigned kMinU = 0x007FFFFFu;  // monotone-mapped -inf
}

// Order-preserving float->uint map (for atomicMax / radix-select on floats).
__device__ __forceinline__ unsigned fmap(float f) {
  unsigned u = __float_as_uint(f);
  return (u & 0x80000000u) ? ~u : (u | 0x80000000u);
}
__device__ __forceinline__ float funmap(unsigned m) {
  unsigned u = (m & 0x80000000u) ? (m & 0x7FFFFFFFu) : ~m;
  return __uint_as_float(u);
}

// ---------------------------------------------------------------------------
// C[N x 128] = A[N x 128] * B[128 x 128] (+bias per column | +=C if accum).
// grid.x = N/16, block = 256 (8 waves). One wave per 16x16 C tile; the 8
// waves cover all 128 output columns of one 16-row stripe.
// A stripe (16x128 f32, 8 KB) is async-copied to LDS once per block
// (global_load_async_to_lds_b128, ASYNCcnt), then WMMA A-fragments are read
// from LDS. B fragments are read from global (64 KB matrix shared by all
// blocks -> WGP$/L2 broadcast).
// f32 WMMA lane layout (ISA 7.12.2):
//   A 16x4 : lane(0..15)=row M, half selects K{0,1}/{2,3} across 2 VGPRs
//   B 4x16 : lane%16=col N, half selects K pair, K across 2 VGPRs
//   C 16x16: vgpr r, lane -> (M = r + 8*(lane>>4), N = lane&15)
// ---------------------------------------------------------------------------
__global__ __launch_bounds__(256) void gemm_wmma_128(
    const float* __restrict__ A, const float* __restrict__ Bm,
    const float* __restrict__ bias, float* __restrict__ C,
    int accum)
{
  __shared__ __align__(16) float As[16 * kLdsStride];  // padded 16x128 stripe

  const int tid  = threadIdx.x;
  const int wave = tid >> 5;
  const int lane = tid & 31;
  const int half = lane >> 4;
  const int l16  = lane & 15;
  const int row0 = blockIdx.x << 4;
  const int col0 = wave << 4;

  // Async copy A stripe: wave w copies rows w and w+8; each lane moves 16 B.
  // LDS generic pointers carry the LDS byte offset in addr[31:0] (ISA 10.2).
  {
    const unsigned lds0 = (unsigned)(size_t)&As[0];
    const unsigned long long g0 =
        (unsigned long long)(const void*)(A + (size_t)row0 * kF);
#pragma unroll
    for (int it = 0; it < 2; ++it) {
      const int r = wave + it * 8;
      unsigned loff = lds0 + (unsigned)(r * kLdsStride * 4 + lane * 16);
      unsigned long long gaddr = g0 + (unsigned long long)(r * kF * 4 + lane * 16);
      asm volatile("global_load_async_to_lds_b128 %0, %1, off"
                   :: "v"(loff), "v"(gaddr)
                   : "memory");
    }
    asm volatile("s_wait_asynccnt 0x0" ::: "memory");
  }
  __syncthreads();

  v8f acc;
  if (accum) {
#pragma unroll
    for (int r = 0; r < 8; ++r)
      acc[r] = C[(size_t)(row0 + r + 8 * half) * kH + col0 + l16];
  } else {
    float bv = bias ? bias[col0 + l16] : 0.0f;
#pragma unroll
    for (int r = 0; r < 8; ++r) acc[r] = bv;
  }

#if __has_builtin(__builtin_amdgcn_wmma_f32_16x16x4_f32)
  const float* arow = &As[l16 * kLdsStride + 2 * half];   // LDS (ds_load_b64)
  const float* bcol = Bm + (size_t)(2 * half) * kH + col0 + l16;
  for (int k = 0; k < kF; k += 4) {
    v2f a = *(const v2f*)(arow + k);
    v2f b;
    b.x = bcol[(size_t)k * kH];
    b.y = bcol[(size_t)k * kH + kH];
    acc = __builtin_amdgcn_wmma_f32_16x16x4_f32(
        false, a, false, b, (short)0, acc, false, false);
  }
#else
  // Host pass / builtin-less fallback: scalar FMA (keeps both passes legal).
  for (int k = 0; k < kF; ++k) {
    float b = Bm[(size_t)k * kH + col0 + l16];
#pragma unroll
    for (int r = 0; r < 8; ++r)
      acc[r] += As[(r + 8 * half) * kLdsStride + k] * b;
  }
#endif

#pragma unroll
  for (int r = 0; r < 8; ++r)
    C[(size_t)(row0 + r + 8 * half) * kH + col0 + l16] = acc[r];
}

// ---------------------------------------------------------------------------
// Two row-dot-products per node: o1[n]=h[n].w1, o2[n]=h[n].w2.
// One wave per node; grid = N/8, block 256.
// ---------------------------------------------------------------------------
__global__ __launch_bounds__(256) void rowdot2(
    const float* __restrict__ h, const float* __restrict__ w1,
    const float* __restrict__ w2, float* __restrict__ o1,
    float* __restrict__ o2)
{
  const int node = blockIdx.x * 8 + (threadIdx.x >> 5);
  const int lane = threadIdx.x & 31;
  float s1 = 0.f, s2 = 0.f;
  for (int c = lane; c < kH; c += 32) {
    float v = h[(size_t)node * kH + c];
    s1 += v * w1[c];
    s2 += v * w2[c];
  }
#pragma unroll
  for (int off = 16; off; off >>= 1) {
    s1 += __shfl_xor(s1, off);
    s2 += __shfl_xor(s2, off);
  }
  if (lane == 0) { o1[node] = s1; o2[node] = s2; }
}

// ---------------------------------------------------------------------------
// GAT(max-aggr) per node: softmax over 8 in-edges + self loop, then
// per-channel max of alpha-weighted source rows, + b_gat. Wave per node.
// ---------------------------------------------------------------------------
__global__ __launch_bounds__(256) void gat_agg(
    const float* __restrict__ h, const int* __restrict__ src,
    const float* __restrict__ a_s, const float* __restrict__ a_d,
    const float* __restrict__ b_gat, float* __restrict__ out)
{
  const int node = blockIdx.x * 8 + (threadIdx.x >> 5);
  const int lane = threadIdx.x & 31;
  int s[kDeg + 1];
#pragma unroll
  for (int i = 0; i < kDeg; ++i) s[i] = src[node * kDeg + i];
  s[kDeg] = node;  // self loop
  const float ad = a_d[node];
  float al[kDeg + 1], m = -INFINITY;
#pragma unroll
  for (int i = 0; i <= kDeg; ++i) {
    float x = a_s[s[i]] + ad;
    al[i] = (x >= 0.f) ? x : 0.2f * x;   // leaky_relu(0.2)
    m = fmaxf(m, al[i]);
  }
  float den = 0.f;
#pragma unroll
  for (int i = 0; i <= kDeg; ++i) { al[i] = expf(al[i] - m); den += al[i]; }
  const float inv = 1.f / den;
#pragma unroll
  for (int i = 0; i <= kDeg; ++i) al[i] *= inv;
  for (int c = lane; c < kH; c += 32) {
    float mx = -INFINITY;
#pragma unroll
    for (int i = 0; i <= kDeg; ++i)
      mx = fmaxf(mx, h[(size_t)s[i] * kH + c] * al[i]);
    out[(size_t)node * kH + c] = mx + b_gat[c];
  }
}

// SAGE max-aggregation over the 8 in-edges. Wave per node.
__global__ __launch_bounds__(256) void sage_agg(
    const float* __restrict__ h, const int* __restrict__ src,
    float* __restrict__ agg)
{
  const int node = blockIdx.x * 8 + (threadIdx.x >> 5);
  const int lane = threadIdx.x & 31;
  int s[kDeg];
#pragma unroll
  for (int i = 0; i < kDeg; ++i) s[i] = src[node * kDeg + i];
  for (int c = lane; c < kH; c += 32) {
    float mx = -INFINITY;
#pragma unroll
    for (int i = 0; i < kDeg; ++i) mx = fmaxf(mx, h[(size_t)s[i] * kH + c]);
    if (!isfinite(mx)) mx = 0.f;
    agg[(size_t)node * kH + c] = mx;
  }
}

// BatchNorm pass 1: per-channel partial sums; 256 blocks x 256 threads.
__global__ __launch_bounds__(256) void bn_stats(
    const float* __restrict__ x, float* __restrict__ sums,
    float* __restrict__ sumsq)
{
  const int c = threadIdx.x & (kH - 1);
  float s = 0.f, ss = 0.f;
  for (int r = blockIdx.x * 2 + (threadIdx.x >> 7); r < kN; r += gridDim.x * 2) {
    float v = x[(size_t)r * kH + c];
    s += v; ss += v * v;
  }
  atomicAdd(&sums[c], s);
  atomicAdd(&sumsq[c], ss);
}

// BatchNorm pass 2 (+PReLU), in place. grid = N*H/256.
__global__ __launch_bounds__(256) void bn_apply(
    float* __restrict__ x, const float* __restrict__ sums,
    const float* __restrict__ sumsq, const float* __restrict__ g,
    const float* __restrict__ b, const float* __restrict__ prelu)
{
  const int idx = blockIdx.x * 256 + threadIdx.x;
  const int c = idx & (kH - 1);
  const float mean = sums[c] * (1.0f / kN);
  const float var  = sumsq[c] * (1.0f / kN) - mean * mean;
  float v = (x[idx] - mean) * rsqrtf(var + kEps) * g[c] + b[c];
  x[idx] = (v >= 0.f) ? v : prelu[0] * v;
}

// SAGPool score: score[n] = sum_{in-edges} p[src] + q[n] + b_pool.
__global__ __launch_bounds__(256) void score_k(
    const float* __restrict__ p, const float* __restrict__ q,
    const int* __restrict__ src, const float* __restrict__ b_pool,
    float* __restrict__ sc, unsigned* __restrict__ scu)
{
  const int n = blockIdx.x * 256 + threadIdx.x;
  float s = q[n] + b_pool[0];
#pragma unroll
  for (int i = 0; i < kDeg; ++i) s += p[src[n * kDeg + i]];
  sc[n] = s;
  scu[n] = fmap(s);
}

// Per-graph K-th largest via MSB radix-select on mapped scores.
// One block (256 thr) per graph; 1024 scores in LDS.
__global__ __launch_bounds__(256) void select_topk(
    const unsigned* __restrict__ u, unsigned* __restrict__ thresh)
{
  __shared__ unsigned sdat[kNPer];
  __shared__ unsigned cred[256];
  __shared__ unsigned pref;
  const int g = blockIdx.x;
  for (int i = threadIdx.x; i < kNPer; i += 256) sdat[i] = u[g * kNPer + i];
  if (threadIdx.x == 0) pref = 0u;
  __syncthreads();
  for (int bit = 31; bit >= 0; --bit) {
    const unsigned test = pref | (1u << bit);
    unsigned c = 0;
    for (int i = threadIdx.x; i < kNPer; i += 256) c += (sdat[i] >= test);
    cred[threadIdx.x] = c;
    __syncthreads();
    for (int st = 128; st; st >>= 1) {
      if (threadIdx.x < st) cred[threadIdx.x] += cred[threadIdx.x + st];
      __syncthreads();
    }
    if (threadIdx.x == 0 && cred[0] >= (unsigned)kKPool) pref = test;
    __syncthreads();
  }
  if (threadIdx.x == 0) thresh[g] = pref;  // K-th largest mapped score
}

// Masked global-max readout of h2*tanh(score) over selected nodes.
// LDS ds_max_u32 pre-reduction, then one global atomicMax per channel/block.
__global__ __launch_bounds__(256) void readout_max(
    const float* __restrict__ h2, const float* __restrict__ sc,
    const unsigned* __restrict__ u, const unsigned* __restrict__ thr,
    unsigned* __restrict__ embed)
{
  __shared__ unsigned sm[kH];
  if (threadIdx.x < kH) sm[threadIdx.x] = kMinU;
  __syncthreads();
  const int node = blockIdx.x * 8 + (threadIdx.x >> 5);
  const int lane = threadIdx.x & 31;
  const int g = node / kNPer;  // 8 consecutive nodes share a graph
  if (u[node] >= thr[g]) {
    const float t = tanhf(sc[node]);
    for (int c = lane; c < kH; c += 32)
      atomicMax(&sm[c], fmap(h2[(size_t)node * kH + c] * t));
  }
  __syncthreads();
  if (threadIdx.x < kH) atomicMax(&embed[threadIdx.x], sm[threadIdx.x]);
}

// Head MLP: pool[258] -> PReLU(linear 258x128) -> dot with W2 -> exp.
__global__ __launch_bounds__(128) void head_kernel(
    const unsigned* __restrict__ eA, const unsigned* __restrict__ eG,
    const float* __restrict__ addf, const float* __restrict__ W1,
    const float* __restrict__ b1, const float* __restrict__ prelu,
    const float* __restrict__ W2, const float* __restrict__ b2,
    float* __restrict__ out)
{
  __shared__ float pool[2 * kH + 2];
  __shared__ float red[kH];
  const int t = threadIdx.x;  // 128
  pool[t]      = funmap(eA[t]);
  pool[kH + t] = funmap(eG[t]);
  if (t < 2) pool[2 * kH + t] = addf[t];
  __syncthreads();
  float s = b1[t];
  for (int i = 0; i < 2 * kH + 2; ++i) s += pool[i] * W1[i * kH + t];
  const float a = prelu[0];
  const float x1 = (s >= 0.f) ? s : a * s;
  red[t] = x1 * W2[t];
  __syncthreads();
  for (int st = 64; st; st >>= 1) {
    if (t < st) red[t] += red[t + st];
    __syncthreads();
  }
  if (t == 0) out[0] = expf(red[0] + b2[0]);
}

__global__ void zero_f32(float* p, int n) {
  int i = blockIdx.x * blockDim.x + threadIdx.x;
  if (i < n) p[i] = 0.f;
}
__global__ void fill_u32(unsigned* p, int n, unsigned v) {
  int i = blockIdx.x * blockDim.x + threadIdx.x;
  if (i < n) p[i] = v;
}

// ---------------------------------------------------------------------------
// Host-side orchestration.
// ---------------------------------------------------------------------------
namespace {
struct Ws {
  float *hbuf, *t1, *t2, *a_s, *a_d, *scoref;
  unsigned* scoreu;
  float *sums;   // sums[kH] followed by sumsq[kH] (contiguous)
  unsigned *thresh, *embedA, *embedG;
};

void run_branch(const float* X, const int* src, void* const* P,
                const Ws& w, unsigned* embed, hipStream_t s)
{
  const float* W_gat   = (const float*)P[0];
  const float* att_src = (const float*)P[1];
  const float* att_dst = (const float*)P[2];
  const float* b_gat   = (const float*)P[3];
  const float* bn1_g   = (const float*)P[4];
  const float* bn1_b   = (const float*)P[5];
  const float* prelu1  = (const float*)P[6];
  const float* Wl      = (const float*)P[7];
  const float* bl      = (const float*)P[8];
  const float* Wr      = (const float*)P[9];
  const float* bn2_g   = (const float*)P[10];
  const float* bn2_b   = (const float*)P[11];
  const float* prelu2  = (const float*)P[12];
  const float* W_rel   = (const float*)P[13];
  const float* W_root  = (const float*)P[14];
  const float* b_pool  = (const float*)P[15];
  float* sumsq = w.sums + kH;

  gemm_wmma_128<<<kN / 16, 256, 0, s>>>(X, W_gat, nullptr, w.hbuf, 0);
  rowdot2<<<kN / 8, 256, 0, s>>>(w.hbuf, att_src, att_dst, w.a_s, w.a_d);
  gat_agg<<<kN / 8, 256, 0, s>>>(w.hbuf, src, w.a_s, w.a_d, b_gat, w.t1);
  zero_f32<<<1, 256, 0, s>>>(w.sums, 2 * kH);
  bn_stats<<<256, 256, 0, s>>>(w.t1, w.sums, sumsq);
  bn_apply<<<kN * kH / 256, 256, 0, s>>>(w.t1, w.sums, sumsq, bn1_g, bn1_b, prelu1);
  sage_agg<<<kN / 8, 256, 0, s>>>(w.t1, src, w.hbuf);
  gemm_wmma_128<<<kN / 16, 256, 0, s>>>(w.hbuf, Wl, bl, w.t2, 0);
  gemm_wmma_128<<<kN / 16, 256, 0, s>>>(w.t1, Wr, nullptr, w.t2, 1);
  zero_f32<<<1, 256, 0, s>>>(w.sums, 2 * kH);
  bn_stats<<<256, 256, 0, s>>>(w.t2, w.sums, sumsq);
  bn_apply<<<kN * kH / 256, 256, 0, s>>>(w.t2, w.sums, sumsq, bn2_g, bn2_b, prelu2);
  rowdot2<<<kN / 8, 256, 0, s>>>(w.t2, W_rel, W_root, w.a_s, w.a_d);  // p, q
  score_k<<<kN / 256, 256, 0, s>>>(w.a_s, w.a_d, src, b_pool, w.scoref, w.scoreu);
  select_topk<<<kB, 256, 0, s>>>(w.scoreu, w.thresh);
  readout_max<<<kN / 8, 256, 0, s>>>(w.t2, w.scoref, w.scoreu, w.thresh, embed);
}
}  // namespace

extern "C" void kernel_launch(void* const* d_in, const int* in_sizes, int n_in,
                              void* d_out, int out_size, void* d_ws, size_t ws_size,
                              hipStream_t stream)
{
  (void)in_sizes; (void)n_in; (void)out_size; (void)ws_size;
  // Input layout (setup_inputs dict order, dicts flattened in insertion order):
  // 0:Abatch_X 1:Abatch_edge_index 2:Abatch_batch 3:Gbatch_X 4:Gbatch_edge_index
  // 5:Gbatch_batch 6:add_features 7..22:A_params 23..38:G_params 39..43:head
  char* p = (char*)d_ws;
  auto alloc = [&](size_t bytes) {
    void* r = (void*)p;
    p += (bytes + 255) & ~(size_t)255;
    return r;
  };
  Ws w;
  w.hbuf   = (float*)alloc((size_t)kN * kH * 4);
  w.t1     = (float*)alloc((size_t)kN * kH * 4);
  w.t2     = (float*)alloc((size_t)kN * kH * 4);
  w.a_s    = (float*)alloc((size_t)kN * 4);
  w.a_d    = (float*)alloc((size_t)kN * 4);
  w.scoref = (float*)alloc((size_t)kN * 4);
  w.scoreu = (unsigned*)alloc((size_t)kN * 4);
  w.sums   = (float*)alloc(2 * kH * 4);
  w.thresh = (unsigned*)alloc(kB * 4);
  w.embedA = (unsigned*)alloc(2 * kH * 4);  // embedA + embedG contiguous
  w.embedG = w.embedA + kH;

  fill_u32<<<1, 256, 0, stream>>>(w.embedA, 2 * kH, kMinU);

  run_branch((const float*)d_in[0], (const int*)d_in[1], d_in + 7,  w, w.embedA, stream);
  run_branch((const float*)d_in[3], (const int*)d_in[4], d_in + 23, w, w.embedG, stream);

  head_kernel<<<1, 128, 0, stream>>>(
      w.embedA, w.embedG, (const float*)d_in[6],
      (const float*)d_in[39], (const float*)d_in[40], (const float*)d_in[41],
      (const float*)d_in[42], (const float*)d_in[43], (float*)d_out);
}